// StructuredAttention_74251394613581
// MI455X (gfx1250) — compile-verified
//
#include <hip/hip_runtime.h>
#include <math.h>

#define B 64
#define N 512
#define D 768

typedef __attribute__((ext_vector_type(2))) float v2f;
typedef __attribute__((ext_vector_type(8))) float v8f;

// ---------------------------------------------------------------------------
// K1: root logits -> R = exp(max(x.w + b - 50*m, -40)).  One wave per row.
// ---------------------------------------------------------------------------
__global__ __launch_bounds__(256) void root_kernel(
    const float* __restrict__ x, const float* __restrict__ mask,
    const float* __restrict__ w_root, const float* __restrict__ b_root,
    float* __restrict__ R) {
  int wave = threadIdx.x >> 5;
  int lane = threadIdx.x & 31;
  int row  = blockIdx.x * 8 + wave;          // [0, B*N)
  int b = row / N, n = row % N;
  const float* xr = x + (size_t)row * D;
  float s = 0.f;
  for (int c = lane; c < D; c += 32) s += xr[c] * w_root[c];
  for (int off = 16; off > 0; off >>= 1) s += __shfl_xor(s, off, 32);
  if (lane == 0) {
    float m = (mask[b * N + n] != 0.f) ? 1.f : 0.f;
    float root = fmaxf(s + b_root[0] - m * 50.f, -40.f);
    R[row] = expf(root);
  }
}

// ---------------------------------------------------------------------------
// K2: LL = -A  (off-diagonal part; diagonal fixed later).
// ---------------------------------------------------------------------------
__global__ __launch_bounds__(256) void build_LL_kernel(
    const float* __restrict__ adj, const float* __restrict__ mask,
    float* __restrict__ LL) {
  size_t idx = (size_t)blockIdx.x * 256 + threadIdx.x;   // < B*N*N
  int b = (int)(idx / ((size_t)N * N));
  int r = (int)(idx % ((size_t)N * N));
  int i = r / N, j = r % N;
  float mi = (mask[b * N + i] != 0.f) ? 1.f : 0.f;
  float mj = (mask[b * N + j] != 0.f) ? 1.f : 0.f;
  float sc = fmaxf(adj[idx] - 50.f * mi - 50.f * mj, -40.f);
  LL[idx] = -expf(sc);
}

// ---------------------------------------------------------------------------
// K3: diagonal fixup: LL[j,j] = L[j] + R[j] - A[j,j], with L[j]=colsum(A).
// Current LL holds -A, so colsum(LL) = -L and LLcur[j,j] = -A[j,j].
// ---------------------------------------------------------------------------
__global__ __launch_bounds__(256) void fix_diag_kernel(
    float* __restrict__ LL, const float* __restrict__ R) {
  int t = blockIdx.x * 256 + threadIdx.x;    // < B*N
  int b = t / N, j = t % N;
  const float* col = LL + (size_t)b * N * N + j;
  float s = 0.f;
  for (int i = 0; i < N; ++i) s += col[(size_t)i * N];
  size_t dix = (size_t)b * N * N + (size_t)j * N + j;
  float cur = LL[dix];                       // = -A[j,j] (included in s)
  LL[dix] = -s + cur + R[t];                 // L + R - A[j,j]
}

// ---------------------------------------------------------------------------
// K4: in-place Gauss-Jordan inversion, pivot-free (LL is column diag-dominant,
// diagonal slack = R > 0).  One workgroup (1024 thr) per batch.
// Key structure: idx = tid + 1024*s  =>  j = tid&511 is CONSTANT per thread,
// i = (tid>>9) + 2*s.  So rowK[j]*p is hoisted out of the hot loop; the inner
// body is one unconditional global load + one FMA + near-uniform selects.
// Working set 1MB/batch (64MB total) is L2-resident (192MB L2).
// ---------------------------------------------------------------------------
__global__ __launch_bounds__(1024) void gj_inverse_kernel(float* __restrict__ LL) {
  __shared__ float rowK[N];
  __shared__ float colK[N];
  float* M = LL + (size_t)blockIdx.x * N * N;
  const int tid  = threadIdx.x;
  const int j    = tid & (N - 1);       // fixed column for this thread
  const int ibase = tid >> 9;           // 0 or 1
  for (int k = 0; k < N; ++k) {
    if (tid < N) rowK[tid] = M[(size_t)k * N + tid];
    else         colK[tid - N] = M[(size_t)(tid - N) * N + k];
    __syncthreads();
    const float p  = 1.f / rowK[k];
    const float rj = rowK[j] * p;       // loop-invariant per step
    const bool  jk = (j == k);
    float* Mp = M + tid;                // element (ibase, j), stride 1024 floats
    #pragma unroll 4
    for (int s = 0; s < (N * N) / 1024; ++s) {
      int i = ibase + 2 * s;
      float mv = Mp[0];                 // unconditional load (L2-resident)
      float ci = colK[i];
      float gen = fmaf(-ci, rj, mv);    // generic elimination update
      float v;
      if (i == k) v = jk ? p : rj;      // wave-uniform condition
      else        v = jk ? (-ci * p) : gen;
      Mp[0] = v;
      Mp += 1024;
    }
    __syncthreads();
  }
}

// ---------------------------------------------------------------------------
// K5: edge marginals d[b,i,j] = A[b,i,j] * (LLinv[j,j] - LLinv[j,i]).
// A recomputed from adj (cheaper than another 64MB buffer).
// ---------------------------------------------------------------------------
__global__ __launch_bounds__(256) void edge_marginal_kernel(
    const float* __restrict__ adj, const float* __restrict__ mask,
    const float* __restrict__ LL, float* __restrict__ dmat) {
  size_t idx = (size_t)blockIdx.x * 256 + threadIdx.x;   // < B*N*N
  int b = (int)(idx / ((size_t)N * N));
  int r = (int)(idx % ((size_t)N * N));
  int i = r / N, j = r % N;
  float mi = (mask[b * N + i] != 0.f) ? 1.f : 0.f;
  float mj = (mask[b * N + j] != 0.f) ? 1.f : 0.f;
  float a  = expf(fmaxf(adj[idx] - 50.f * mi - 50.f * mj, -40.f));
  size_t base = (size_t)b * N * N;
  float diagj = LL[base + (size_t)j * N + j];
  float tji   = LL[base + (size_t)j * N + i];
  dmat[idx] = a * (diagj - tji);
}

// ---------------------------------------------------------------------------
// K6: deterministic BCE root loss in one 1024-thread block.
// ---------------------------------------------------------------------------
__global__ __launch_bounds__(1024) void loss_kernel(
    const float* __restrict__ R, const float* __restrict__ LL,
    const float* __restrict__ roots_label, const int* __restrict__ root_mask,
    float* __restrict__ loss) {
  __shared__ float sdata[1024];
  float acc = 0.f;
  for (int t = threadIdx.x; t < B * N; t += 1024) {
    int b = t / N, n = t % N;
    float diag = LL[(size_t)b * N * N + (size_t)n * N + n];
    float d0 = R[t] * diag;
    float p = fminf(fmaxf(d0, 1e-5f), 1.f - 1e-5f);
    int rm = root_mask[t];
    float label = (rm == 1) ? roots_label[t] : 0.f;
    float bce = -(label * logf(p) + (1.f - label) * logf(1.f - p));
    acc += bce * (float)rm;
  }
  sdata[threadIdx.x] = acc;
  __syncthreads();
  for (int s = 512; s > 0; s >>= 1) {
    if ((int)threadIdx.x < s) sdata[threadIdx.x] += sdata[threadIdx.x + s];
    __syncthreads();
  }
  if (threadIdx.x == 0) *loss = sdata[0] / (float)(B * N);
}

// ---------------------------------------------------------------------------
// K7: context = attn @ x via V_WMMA_F32_16X16X4_F32 (full fp32 precision).
// attn[b,i,k] = maskf[k] * d[b,k,i]  (transpose + key-column zeroing).
// Each wave computes a 16x64 output tile: one A-frag feeds 4 accumulators
// (4 wmma per K-step of 4).  All 8 waves of a block share batch b, so the
// 512-entry 0/1 mask-factor table lives in LDS -> branch-free masking.
//   A 16x4:  lanes 0-15 -> (M=lane, K=k0,k0+1); lanes 16-31 -> K=k0+2,k0+3
//   B 4x16:  VGPR0 = rows K0/K2, VGPR1 = rows K1/K3
//   C/D:     VGPR v = row v (lanes 0-15) / row v+8 (lanes 16-31)
// ---------------------------------------------------------------------------
__global__ __launch_bounds__(256) void context_wmma_kernel(
    const float* __restrict__ dmat, const float* __restrict__ x,
    const float* __restrict__ mask, float* __restrict__ ctx) {
  const int MT = N / 16;            // 32 row tiles
  const int NG = D / 64;            // 12 col groups (64 wide)
  const int BLOCKS_PER_BATCH = (MT * NG) / 8;   // 48
  __shared__ float maskf[N];

  int b = blockIdx.x / BLOCKS_PER_BATCH;
  // 0/1 keep-factors for key columns of this batch
  for (int t = threadIdx.x; t < N; t += 256)
    maskf[t] = (mask[b * N + t] != 0.f) ? 0.f : 1.f;
  __syncthreads();

  int wave = threadIdx.x >> 5;
  int lane = threadIdx.x & 31;
  int wtile = (blockIdx.x % BLOCKS_PER_BATCH) * 8 + wave;  // [0, MT*NG)
  int mt = wtile / NG;
  int ng = wtile % NG;
  int row0 = mt * 16, col0 = ng * 64;
  int half = lane >> 4;
  int lr   = lane & 15;

  const float* dB = dmat + (size_t)b * N * N;
  const float* xB = x    + (size_t)b * N * D;
  int arow = row0 + lr;             // output row this lane supplies for A
  int bcol = col0 + lr;             // base output col this lane supplies

  // streaming pointers: element (K = 2*half + k0, arow/bcol)
  const float* pa = dB + (size_t)(2 * half) * N + arow;
  const float* pb = xB + (size_t)(2 * half) * D + bcol;

  v8f c0 = {0.f,0.f,0.f,0.f,0.f,0.f,0.f,0.f};
  v8f c1 = c0, c2 = c0, c3 = c0;

  for (int k0 = 0; k0 < N; k0 += 4) {
    int ka = k0 + 2 * half;
    float mf0 = maskf[ka];
    float mf1 = maskf[ka + 1];
    v2f afrag = { pa[0] * mf0, pa[N] * mf1 };     // unconditional loads
    v2f b0 = { pb[0],  pb[D] };
    v2f b1 = { pb[16], pb[D + 16] };
    v2f b2 = { pb[32], pb[D + 32] };
    v2f b3 = { pb[48], pb[D + 48] };
    c0 = __builtin_amdgcn_wmma_f32_16x16x4_f32(false, afrag, false, b0, (short)0, c0, false, false);
    c1 = __builtin_amdgcn_wmma_f32_16x16x4_f32(false, afrag, false, b1, (short)0, c1, false, false);
    c2 = __builtin_amdgcn_wmma_f32_16x16x4_f32(false, afrag, false, b2, (short)0, c2, false, false);
    c3 = __builtin_amdgcn_wmma_f32_16x16x4_f32(false, afrag, false, b3, (short)0, c3, false, false);
    pa += 4 * N;
    pb += 4 * D;
  }

  float* out = ctx + (size_t)b * N * D;
  #pragma unroll
  for (int v = 0; v < 8; ++v) {
    int orow = row0 + v + 8 * half;
    float* orow_p = out + (size_t)orow * D + bcol;
    orow_p[0]  = c0[v];
    orow_p[16] = c1[v];
    orow_p[32] = c2[v];
    orow_p[48] = c3[v];
  }
}

// ---------------------------------------------------------------------------
extern "C" void kernel_launch(void* const* d_in, const int* in_sizes, int n_in,
                              void* d_out, int out_size, void* d_ws, size_t ws_size,
                              hipStream_t stream) {
  const float* x           = (const float*)d_in[0];
  const float* adj         = (const float*)d_in[1];
  const float* mask        = (const float*)d_in[2];   // [B,1,1,N] flat = B*N
  const float* roots_label = (const float*)d_in[3];
  const int*   root_mask   = (const int*)d_in[4];
  const float* w_root      = (const float*)d_in[5];
  const float* b_root      = (const float*)d_in[6];

  float* out  = (float*)d_out;
  float* ctx  = out;                                    // B*N*D
  float* dmat = out + (size_t)B * N * D;                // B*N*N
  float* loss = out + (size_t)B * N * D + (size_t)B * N * N;  // 1

  float* ws = (float*)d_ws;
  float* LL = ws;                                       // B*N*N (64 MB)
  float* R  = ws + (size_t)B * N * N;                   // B*N

  root_kernel<<<(B * N) / 8, 256, 0, stream>>>(x, mask, w_root, b_root, R);
  build_LL_kernel<<<(size_t)B * N * N / 256, 256, 0, stream>>>(adj, mask, LL);
  fix_diag_kernel<<<(B * N) / 256, 256, 0, stream>>>(LL, R);
  gj_inverse_kernel<<<B, 1024, 0, stream>>>(LL);
  edge_marginal_kernel<<<(size_t)B * N * N / 256, 256, 0, stream>>>(adj, mask, LL, dmat);
  loss_kernel<<<1, 1024, 0, stream>>>(R, LL, roots_label, root_mask, loss);
  context_wmma_kernel<<<(B * (N / 16) * (D / 64)) / 8, 256, 0, stream>>>(dmat, x, mask, ctx);
}